// LuongAttnDecoder_30872224924158
// MI455X (gfx1250) — compile-verified
//
#include <hip/hip_runtime.h>

#define V_ 32000
#define E_ 1024
#define H_ 1024
#define L_ 2
#define B_ 64
#define SRC_ 128

typedef __attribute__((ext_vector_type(2))) float v2f;
typedef __attribute__((ext_vector_type(8))) float v8f;

__device__ __forceinline__ float sigmoidf_(float x) { return 1.0f / (1.0f + expf(-x)); }
__device__ __forceinline__ int rfl(int v) { return __builtin_amdgcn_readfirstlane(v); }

// One wave computes a 16-wide N-panel x 64 rows over a K segment.
// V_WMMA_F32_16X16X4_F32 lane layout: lanes 0-15 hold {A[m][k],A[m][k+1]},
// lanes 16-31 hold {A[m][k+2],A[m][k+3]}; B mirrors with N across lanes.
// len is always a compile-time constant at call sites -> unrolled by 4,
// scalar loop control, EXEC untouched.
__device__ __forceinline__ void mma_seg(const float* __restrict__ Arow,
                                        const float* __restrict__ Wrow,
                                        int len, int lda,
                                        v8f& c0, v8f& c1, v8f& c2, v8f& c3) {
  const int S = 16 * lda;
#pragma unroll 4
  for (int k = 0; k < len; k += 4) {
    __builtin_prefetch(Wrow + k + 128, 0, 1);   // global_prefetch_b8 on weight stream
    v2f b  = *(const v2f*)(Wrow + k);
    v2f a0 = *(const v2f*)(Arow + k);
    v2f a1 = *(const v2f*)(Arow + S + k);
    v2f a2 = *(const v2f*)(Arow + 2 * S + k);
    v2f a3 = *(const v2f*)(Arow + 3 * S + k);
    c0 = __builtin_amdgcn_wmma_f32_16x16x4_f32(false, a0, false, b, (short)0, c0, false, false);
    c1 = __builtin_amdgcn_wmma_f32_16x16x4_f32(false, a1, false, b, (short)0, c1, false, false);
    c2 = __builtin_amdgcn_wmma_f32_16x16x4_f32(false, a2, false, b, (short)0, c2, false, false);
    c3 = __builtin_amdgcn_wmma_f32_16x16x4_f32(false, a3, false, b, (short)0, c3, false, false);
  }
}

// LSTM gate partials, split-K over combined K=2048 ([x;h0] against [Wih;Whh]).
// part[sp][64][4096]; grid = 32*NS blocks of 256.
template <int NS>
__global__ void lstm_gates_split(const float* __restrict__ x,
                                 const float* __restrict__ hprev,
                                 const float* __restrict__ Wih,
                                 const float* __restrict__ Whh,
                                 float* __restrict__ part) {
  const int wave = rfl(threadIdx.x >> 5);        // scalar: uniform branches, no EXEC play
  const int lane = threadIdx.x & 31;
  const int w = blockIdx.x * 8 + wave;
  const int tileN = w % 256;
  const int sp = w / 256;
  const int half = lane >> 4, r = lane & 15;
  const int n = tileN * 16 + r;
  v8f c0 = {}, c1 = {}, c2 = {}, c3 = {};
  if (NS == 1) {
    mma_seg(x + (size_t)r * E_ + 2 * half, Wih + (size_t)n * E_ + 2 * half, E_, E_, c0, c1, c2, c3);
    mma_seg(hprev + (size_t)r * H_ + 2 * half, Whh + (size_t)n * H_ + 2 * half, H_, H_, c0, c1, c2, c3);
  } else {
    constexpr int SEGS = NS / 2;                 // segments per matrix
    constexpr int LEN = 2048 / NS;               // compile-time segment length
    if (sp < SEGS) {
      const int kb = sp * LEN;
      mma_seg(x + (size_t)r * E_ + 2 * half + kb,
              Wih + (size_t)n * E_ + 2 * half + kb, LEN, E_, c0, c1, c2, c3);
    } else {
      const int kb = (sp - SEGS) * LEN;
      mma_seg(hprev + (size_t)r * H_ + 2 * half + kb,
              Whh + (size_t)n * H_ + 2 * half + kb, LEN, H_, c0, c1, c2, c3);
    }
  }
  const int mb = half * 8;
  float* p = part + (size_t)sp * B_ * 4 * H_;
#pragma unroll
  for (int j = 0; j < 8; ++j) {
    p[(size_t)(mb + j) * (4 * H_) + n]      = c0[j];
    p[(size_t)(16 + mb + j) * (4 * H_) + n] = c1[j];
    p[(size_t)(32 + mb + j) * (4 * H_) + n] = c2[j];
    p[(size_t)(48 + mb + j) * (4 * H_) + n] = c3[j];
  }
}

// Fused: sum split-K partials + biases, LSTM cell activations.
__global__ void lstm_activate_fused(const float* __restrict__ part,
                                    const float* __restrict__ bih,
                                    const float* __restrict__ bhh,
                                    const float* __restrict__ c0v,
                                    float* __restrict__ hn, float* __restrict__ cn,
                                    float* __restrict__ hcur, int nsplit) {
  const int idx = blockIdx.x * blockDim.x + threadIdx.x;   // b*1024 + j
  const int b = idx >> 10, j = idx & 1023;
  float gi = bih[j] + bhh[j];
  float gf = bih[H_ + j] + bhh[H_ + j];
  float gg = bih[2 * H_ + j] + bhh[2 * H_ + j];
  float go = bih[3 * H_ + j] + bhh[3 * H_ + j];
  for (int sp = 0; sp < nsplit; ++sp) {
    const float* p = part + (size_t)sp * B_ * 4 * H_ + (size_t)b * 4 * H_;
    gi += p[j]; gf += p[H_ + j]; gg += p[2 * H_ + j]; go += p[3 * H_ + j];
  }
  const float c = sigmoidf_(gf) * c0v[idx] + sigmoidf_(gi) * tanhf(gg);
  const float h = sigmoidf_(go) * tanhf(c);
  cn[idx] = c; hn[idx] = h; hcur[idx] = h;
}

// Concat projection split-K partials: K=2048, N=1024 fixed.
// part[sp][64][1024]; grid = 8*NS blocks of 256.
template <int NS>
__global__ void concat_gemm_split(const float* __restrict__ A,     // [64,2048]
                                  const float* __restrict__ W,     // [1024,2048]
                                  float* __restrict__ part) {
  const int wave = rfl(threadIdx.x >> 5);
  const int lane = threadIdx.x & 31;
  const int w = blockIdx.x * 8 + wave;
  const int tileN = w % 64;
  const int sp = w / 64;
  const int half = lane >> 4, r = lane & 15;
  const int n = tileN * 16 + r;
  constexpr int K = 2 * H_;
  constexpr int LEN = K / NS;
  const int kb = sp * LEN;
  v8f c0 = {}, c1 = {}, c2 = {}, c3 = {};
  mma_seg(A + (size_t)r * K + 2 * half + kb,
          W + (size_t)n * K + 2 * half + kb, LEN, K, c0, c1, c2, c3);
  const int mb = half * 8;
  float* p = part + (size_t)sp * B_ * H_;
#pragma unroll
  for (int j = 0; j < 8; ++j) {
    p[(size_t)(mb + j) * H_ + n]      = c0[j];
    p[(size_t)(16 + mb + j) * H_ + n] = c1[j];
    p[(size_t)(32 + mb + j) * H_ + n] = c2[j];
    p[(size_t)(48 + mb + j) * H_ + n] = c3[j];
  }
}

template <int ACT>
__global__ void reduce_bias_act(const float* __restrict__ part,
                                const float* __restrict__ bias,
                                float* __restrict__ C, int N, int nsplit) {
  const int idx = blockIdx.x * blockDim.x + threadIdx.x;   // m*N + n
  const int n = idx % N;
  float s = bias[n];
  for (int sp = 0; sp < nsplit; ++sp) s += part[(size_t)sp * B_ * N + idx];
  C[idx] = (ACT == 1) ? tanhf(s) : s;
}

// Vocab projection: each wave owns TWO 16-wide N-tiles (shared A loads),
// streams out_W exactly once with non-temporal loads. grid = 125 blocks of 256.
__global__ void out_gemm_wmma(const float* __restrict__ A,     // [64,1024]
                              const float* __restrict__ W,     // [32000,1024]
                              const float* __restrict__ bias,
                              float* __restrict__ C) {
  const int wave = rfl(threadIdx.x >> 5);
  const int lane = threadIdx.x & 31;
  const int tp = blockIdx.x * 8 + wave;          // tile pair 0..999
  const int half = lane >> 4, r = lane & 15;
  const int n0 = tp * 32 + r, n1 = n0 + 16;
  const float* W0 = W + (size_t)n0 * H_ + 2 * half;
  const float* W1 = W + (size_t)n1 * H_ + 2 * half;
  const float* A0 = A + (size_t)r * H_ + 2 * half;
  v8f z = {};
  v8f acc[8] = {z, z, z, z, z, z, z, z};
#pragma unroll 4
  for (int k = 0; k < H_; k += 4) {
    __builtin_prefetch(W0 + k + 128, 0, 0);
    __builtin_prefetch(W1 + k + 128, 0, 0);
    v2f b0 = __builtin_nontemporal_load((const v2f*)(W0 + k));  // read-once stream: keep L2 for reuse data
    v2f b1 = __builtin_nontemporal_load((const v2f*)(W1 + k));
    v2f a0 = *(const v2f*)(A0 + k);
    v2f a1 = *(const v2f*)(A0 + 16 * H_ + k);
    v2f a2 = *(const v2f*)(A0 + 32 * H_ + k);
    v2f a3 = *(const v2f*)(A0 + 48 * H_ + k);
    acc[0] = __builtin_amdgcn_wmma_f32_16x16x4_f32(false, a0, false, b0, (short)0, acc[0], false, false);
    acc[1] = __builtin_amdgcn_wmma_f32_16x16x4_f32(false, a1, false, b0, (short)0, acc[1], false, false);
    acc[2] = __builtin_amdgcn_wmma_f32_16x16x4_f32(false, a2, false, b0, (short)0, acc[2], false, false);
    acc[3] = __builtin_amdgcn_wmma_f32_16x16x4_f32(false, a3, false, b0, (short)0, acc[3], false, false);
    acc[4] = __builtin_amdgcn_wmma_f32_16x16x4_f32(false, a0, false, b1, (short)0, acc[4], false, false);
    acc[5] = __builtin_amdgcn_wmma_f32_16x16x4_f32(false, a1, false, b1, (short)0, acc[5], false, false);
    acc[6] = __builtin_amdgcn_wmma_f32_16x16x4_f32(false, a2, false, b1, (short)0, acc[6], false, false);
    acc[7] = __builtin_amdgcn_wmma_f32_16x16x4_f32(false, a3, false, b1, (short)0, acc[7], false, false);
  }
  const int mb = half * 8;
  const float bv0 = bias[n0], bv1 = bias[n1];
#pragma unroll
  for (int t = 0; t < 4; ++t) {
#pragma unroll
    for (int j = 0; j < 8; ++j) {
      const int m = t * 16 + mb + j;
      C[(size_t)m * V_ + n0] = acc[t][j] + bv0;
      C[(size_t)m * V_ + n1] = acc[4 + t][j] + bv1;
    }
  }
}

__global__ void embed_kernel(const int* __restrict__ tokens,
                             const float* __restrict__ emb,
                             float* __restrict__ x) {
  const int b = blockIdx.x;
  const int row = tokens[b];
  const float4* src = (const float4*)(emb + (size_t)row * E_);
  float4* dst = (float4*)(x + (size_t)b * E_);
  dst[threadIdx.x] = src[threadIdx.x];
}

// u2 partials over 16 h-chunks x 4 j-blocks -> 64 blocks
__global__ void attn_u_part_kernel(const float* __restrict__ attn_v,
                                   const float* __restrict__ attn_W,
                                   float* __restrict__ u2part) {
  const int jb = blockIdx.x & 3, hc = blockIdx.x >> 2;
  const int j = jb * 256 + threadIdx.x;
  float s = 0.0f;
  const int h0 = hc * 64;
  for (int h = h0; h < h0 + 64; ++h) s += attn_v[h] * attn_W[(size_t)h * (2 * H_) + H_ + j];
  u2part[hc * H_ + j] = s;
}

__global__ void attn_u_reduce_kernel(const float* __restrict__ u2part,
                                     float* __restrict__ u2) {
  const int j = blockIdx.x * blockDim.x + threadIdx.x;
  float s = 0.0f;
  for (int hc = 0; hc < 16; ++hc) s += u2part[hc * H_ + j];
  u2[j] = s;
}

// scores[b][l] = u2 . enc[l,b,:]  -- one wave per (b,l); grid 64*16 blocks.
__global__ void attn_scores_kernel(const float* __restrict__ enc,
                                   const float* __restrict__ u2,
                                   float* __restrict__ scores) {
  const int b = blockIdx.x >> 4;
  const int lc = blockIdx.x & 15;
  const int wave = threadIdx.x >> 5, lane = threadIdx.x & 31;
  const int l = lc * 8 + wave;
  const float* e = enc + ((size_t)l * B_ + b) * H_;
  float s = 0.0f;
  for (int h = lane; h < H_; h += 32) s += u2[h] * e[h];
#pragma unroll
  for (int off = 16; off > 0; off >>= 1) s += __shfl_xor(s, off, 32);
  if (lane == 0) scores[b * SRC_ + l] = s;
}

__global__ void attn_softmax_kernel(float* __restrict__ scores) {
  __shared__ float sc[SRC_];
  __shared__ float stats[2];
  const int b = blockIdx.x;
  for (int l = threadIdx.x; l < SRC_; l += blockDim.x) sc[l] = scores[b * SRC_ + l];
  __syncthreads();
  if (threadIdx.x == 0) {
    float mx = sc[0];
    for (int l = 1; l < SRC_; ++l) mx = fmaxf(mx, sc[l]);
    float sum = 0.0f;
    for (int l = 0; l < SRC_; ++l) sum += expf(sc[l] - mx);
    stats[0] = mx; stats[1] = 1.0f / sum;
  }
  __syncthreads();
  for (int l = threadIdx.x; l < SRC_; l += blockDim.x)
    scores[b * SRC_ + l] = expf(sc[l] - stats[0]) * stats[1];
}

// context + concat_in assembly; grid 64*4 blocks, 256 h-cols each.
__global__ void attn_context_kernel(const float* __restrict__ enc,
                                    const float* __restrict__ scores,
                                    const float* __restrict__ htop,
                                    float* __restrict__ A2) {
  __shared__ float wgt[SRC_];
  const int b = blockIdx.x >> 2;
  const int hc = blockIdx.x & 3;
  for (int l = threadIdx.x; l < SRC_; l += blockDim.x) wgt[l] = scores[b * SRC_ + l];
  __syncthreads();
  const int h = hc * 256 + threadIdx.x;
  float acc = 0.0f;
  for (int l = 0; l < SRC_; ++l) acc += wgt[l] * enc[((size_t)l * B_ + b) * H_ + h];
  A2[(size_t)b * (2 * H_) + h] = acc;
  A2[(size_t)b * (2 * H_) + H_ + h] = htop[(size_t)b * H_ + h];
}

__global__ void logsoftmax_kernel(float* __restrict__ logp) {
  __shared__ float red[256];
  const int b = blockIdx.x, t = threadIdx.x;
  float* row = logp + (size_t)b * V_;
  float mx = -3.4e38f;
  for (int i = t; i < V_; i += 256) mx = fmaxf(mx, row[i]);
  red[t] = mx;
  __syncthreads();
  for (int s = 128; s > 0; s >>= 1) { if (t < s) red[t] = fmaxf(red[t], red[t + s]); __syncthreads(); }
  mx = red[0];
  __syncthreads();
  float sum = 0.0f;
  for (int i = t; i < V_; i += 256) sum += expf(row[i] - mx);
  red[t] = sum;
  __syncthreads();
  for (int s = 128; s > 0; s >>= 1) { if (t < s) red[t] += red[t + s]; __syncthreads(); }
  const float lse = mx + logf(red[0]);
  for (int i = t; i < V_; i += 256) row[i] = row[i] - lse;
}

extern "C" void kernel_launch(void* const* d_in, const int* in_sizes, int n_in,
                              void* d_out, int out_size, void* d_ws, size_t ws_size,
                              hipStream_t stream) {
  const int*   tokens = (const int*)d_in[0];
  const float* h0     = (const float*)d_in[1];
  const float* c0     = (const float*)d_in[2];
  const float* enc    = (const float*)d_in[3];
  const float* emb    = (const float*)d_in[4];
  const float* Wih    = (const float*)d_in[5];
  const float* Whh    = (const float*)d_in[6];
  const float* bih    = (const float*)d_in[7];
  const float* bhh    = (const float*)d_in[8];
  const float* attnW  = (const float*)d_in[9];
  /* d_in[10] attn_b: constant over softmax axis -> cancels */
  const float* attnv  = (const float*)d_in[11];
  const float* concW  = (const float*)d_in[12];
  const float* concb  = (const float*)d_in[13];
  const float* outW   = (const float*)d_in[14];
  const float* outb   = (const float*)d_in[15];

  float* out  = (float*)d_out;
  float* logp = out;                              // [64, 32000]
  float* hn   = out + (size_t)B_ * V_;            // [2, 64, 1024]
  float* cn   = hn + (size_t)L_ * B_ * H_;        // [2, 64, 1024]

  // Choose split-K factor by available scratch (deterministic per harness).
  const size_t fixed = 353280;                    // floats
  int NS = 4;
  if (ws_size < (fixed + 4u * 327680u) * sizeof(float)) NS = 2;
  if (ws_size < (fixed + 2u * 327680u) * sizeof(float)) NS = 1;

  float* ws     = (float*)d_ws;
  float* x      = ws;                               // [64,1024]
  float* partG  = x + B_ * E_;                      // [NS,64,4096]
  float* hcur   = partG + (size_t)NS * B_ * 4 * H_; // [64,1024]
  float* u2part = hcur + B_ * H_;                   // [16,1024]
  float* u2     = u2part + 16 * H_;                 // [1024]
  float* scores = u2 + H_;                          // [64,128]
  float* A2     = scores + B_ * SRC_;               // [64,2048]
  float* partC  = A2 + B_ * 2 * H_;                 // [NS,64,1024]
  float* cout   = partC + (size_t)NS * B_ * H_;     // [64,1024]

  embed_kernel<<<B_, 256, 0, stream>>>(tokens, emb, x);
  attn_u_part_kernel<<<64, 256, 0, stream>>>(attnv, attnW, u2part);
  attn_u_reduce_kernel<<<H_ / 256, 256, 0, stream>>>(u2part, u2);

  const float* layer_in = x;
  for (int l = 0; l < L_; ++l) {
    const float* hp = h0 + (size_t)l * B_ * H_;
    const float* wi = Wih + (size_t)l * 4 * H_ * E_;
    const float* wh = Whh + (size_t)l * 4 * H_ * H_;
    if (NS == 4)
      lstm_gates_split<4><<<128, 256, 0, stream>>>(layer_in, hp, wi, wh, partG);
    else if (NS == 2)
      lstm_gates_split<2><<<64, 256, 0, stream>>>(layer_in, hp, wi, wh, partG);
    else
      lstm_gates_split<1><<<32, 256, 0, stream>>>(layer_in, hp, wi, wh, partG);
    lstm_activate_fused<<<(B_ * H_) / 256, 256, 0, stream>>>(
        partG, bih + (size_t)l * 4 * H_, bhh + (size_t)l * 4 * H_,
        c0 + (size_t)l * B_ * H_,
        hn + (size_t)l * B_ * H_, cn + (size_t)l * B_ * H_, hcur, NS);
    layer_in = hcur;
  }

  attn_scores_kernel<<<B_ * 16, 256, 0, stream>>>(enc, u2, scores);
  attn_softmax_kernel<<<B_, 256, 0, stream>>>(scores);
  attn_context_kernel<<<B_ * 4, 256, 0, stream>>>(enc, scores, hcur, A2);

  if (NS == 4)
    concat_gemm_split<4><<<32, 256, 0, stream>>>(A2, concW, partC);
  else if (NS == 2)
    concat_gemm_split<2><<<16, 256, 0, stream>>>(A2, concW, partC);
  else
    concat_gemm_split<1><<<8, 256, 0, stream>>>(A2, concW, partC);
  reduce_bias_act<1><<<(B_ * H_) / 256, 256, 0, stream>>>(partC, concb, cout, H_, NS);

  out_gemm_wmma<<<(V_ / 16) / 16, 256, 0, stream>>>(cout, outW, outb, logp);
  logsoftmax_kernel<<<B_, 256, 0, stream>>>(logp);
}